// PathologyAwareRhythmicStateMamba_82987358094114
// MI455X (gfx1250) — compile-verified
//
#include <hip/hip_runtime.h>
#include <hip/hip_bf16.h>
#include <math.h>
#include <stdint.h>

typedef __attribute__((ext_vector_type(2))) float        v2f;
typedef __attribute__((ext_vector_type(8))) float        v8f;
typedef __attribute__((ext_vector_type(4))) unsigned int v4u;
typedef __attribute__((ext_vector_type(4))) int          v4i;
typedef __attribute__((ext_vector_type(8))) int          v8i;

constexpr int Bv     = 2;
constexpr int Sv     = 4096;
constexpr int Dv     = 128;
constexpr int INNERv = 256;
constexpr int DRv    = 8;
constexpr int Nv     = 16;
constexpr int Mrows  = Bv * Sv;   // 8192 tokens

__device__ __forceinline__ float siluf(float x) { return x / (1.0f + expf(-x)); }
__device__ __forceinline__ float sigmf(float x) { return 1.0f / (1.0f + expf(-x)); }

// ---------------------------------------------------------------------------
// fp32 WMMA 16x16x4, register-blocked 16x(NT*16) per wave.
// acc[i] = A[m0:+16, 0:K] * W[n0+16i:+16, 0:K]^T ; A fragment reused NT times.
// W may point to global memory or LDS (inlined -> ds_load path).
// ---------------------------------------------------------------------------
template <int NT>
__device__ __forceinline__ void wmma_tileN(const float* __restrict__ A, int lda,
                                           const float* W, int ldw,
                                           int m0, int n0, int K, v8f acc[NT]) {
  const int lane = threadIdx.x & 31;
  const int row  = lane & 15;
  const int koff = (lane >> 4) << 1;
  const float* a = A + (size_t)(m0 + row) * lda + koff;
  const float* w = W + (size_t)(n0 + row) * ldw + koff;
  __builtin_prefetch(a, 0, 1);          // global_prefetch_b8
  v8f z = {};
  #pragma unroll
  for (int i = 0; i < NT; ++i) acc[i] = z;
  for (int k = 0; k < K; k += 4) {
    v2f av; av.x = a[k]; av.y = a[k + 1];
    #pragma unroll
    for (int i = 0; i < NT; ++i) {
      const float* wi = w + (size_t)(i * 16) * ldw;
      v2f wv; wv.x = wi[k]; wv.y = wi[k + 1];
      acc[i] = __builtin_amdgcn_wmma_f32_16x16x4_f32(false, av, false, wv,
                                                     (short)0, acc[i], false, false);
    }
  }
}

// ---------------------------------------------------------------------------
// Tensor Data Mover: stage a (nrows x rowElems) fp32 2D tile global -> LDS.
// D# packed per CDNA5 ISA ch.8: group0 = {count|flags, lds_addr, gaddr_lo,
// gaddr_hi|type=2}; group1 carries data_size/dims/tile/stride.
// ---------------------------------------------------------------------------
__device__ __forceinline__ void tdm_stage_rows(const float* gsrc, float* lds_dst,
                                               int rowElems, int nrows) {
  unsigned long long ga = (unsigned long long)(uintptr_t)gsrc;
  unsigned int ldsoff   = (unsigned int)(uintptr_t)lds_dst;  // low 32 = LDS byte offset
  v4u g0;
  g0.x = 1u;                                                  // count=1 (valid)
  g0.y = ldsoff;                                              // lds_addr
  g0.z = (unsigned int)ga;                                    // global_addr[31:0]
  g0.w = ((unsigned int)(ga >> 32) & 0x01FFFFFFu) | (2u << 30); // addr[56:32]|type=2
  v8i g1;
  g1[0] = (int)(2u << 16);                                    // data_size = 4B
  g1[1] = (int)((unsigned)(rowElems & 0xFFFF) << 16);         // tensor_dim0 lo16
  g1[2] = (int)((unsigned)(nrows & 0xFFFF) << 16);            // dim0 hi=0 | tensor_dim1 lo16
  g1[3] = (int)((unsigned)(rowElems & 0xFFFF) << 16);         // dim1 hi=0 | tile_dim0
  g1[4] = nrows;                                              // tile_dim1 | tile_dim2=0
  g1[5] = rowElems;                                           // tensor_dim0_stride lo32
  g1[6] = 0;
  g1[7] = 0;
  v4i gz; gz.x = 0; gz.y = 0; gz.z = 0; gz.w = 0;
#if defined(__clang_major__) && (__clang_major__ >= 23)
  v8i g4; g4[0]=0; g4[1]=0; g4[2]=0; g4[3]=0; g4[4]=0; g4[5]=0; g4[6]=0; g4[7]=0;
  __builtin_amdgcn_tensor_load_to_lds(g0, g1, gz, gz, g4, 0);
#else
  __builtin_amdgcn_tensor_load_to_lds(g0, g1, gz, gz, 0);
#endif
  __builtin_amdgcn_s_wait_tensorcnt((short)0);
}

// ---------------------------------------------------------------------------
// LayerNorm over D=128, one wave per row (wave32: 4 elems/lane)
// ---------------------------------------------------------------------------
__global__ void k_ln(const float* in, const float* g, const float* b,
                     float* out, int rows) {
  int wave = blockIdx.x * (blockDim.x >> 5) + (threadIdx.x >> 5);
  if (wave >= rows) return;
  int lane = threadIdx.x & 31;
  const float* r = in + (size_t)wave * Dv;
  float v0[4], s = 0.f, s2 = 0.f;
  #pragma unroll
  for (int i = 0; i < 4; ++i) { v0[i] = r[lane + 32 * i]; s += v0[i]; s2 += v0[i] * v0[i]; }
  #pragma unroll
  for (int off = 16; off; off >>= 1) {
    s  += __shfl_xor(s,  off, 32);
    s2 += __shfl_xor(s2, off, 32);
  }
  float mean = s * (1.f / Dv);
  float var  = fmaxf(s2 * (1.f / Dv) - mean * mean, 0.f);
  float inv  = rsqrtf(var + 1e-5f);
  float* o = out + (size_t)wave * Dv;
  #pragma unroll
  for (int i = 0; i < 4; ++i) {
    int c = lane + 32 * i;
    o[c] = (v0[i] - mean) * inv * g[c] + b[c];
  }
}

// slow = dwconv(x, w_low, k=9 same), fast = x - slow     (layout [B*S, D])
__global__ void k_dwlow(const float* __restrict__ x, const float* __restrict__ w,
                        float* __restrict__ slow, float* __restrict__ fast) {
  int idx = blockIdx.x * blockDim.x + threadIdx.x;
  if (idx >= Mrows * Dv) return;
  int d  = idx & (Dv - 1);
  int ms = idx >> 7;
  int s  = ms & (Sv - 1);
  int bb = ms >> 12;
  float acc = 0.f;
  #pragma unroll
  for (int k = 0; k < 9; ++k) {
    int si = s - 4 + k;
    if (si >= 0 && si < Sv)
      acc += x[(((size_t)(bb * Sv + si)) << 7) + d] * w[d * 9 + k];
  }
  slow[idx] = acc;
  fast[idx] = x[idx] - acc;
}

// multi-scale depthwise convs -> concat 3*D channels  (out [B*S, 384])
__global__ void k_msconv(const float* __restrict__ src,
                         const float* __restrict__ w0, int k0,
                         const float* __restrict__ w1, int k1,
                         const float* __restrict__ w2, int k2,
                         float* __restrict__ out) {
  int idx = blockIdx.x * blockDim.x + threadIdx.x;
  if (idx >= Mrows * 3 * Dv) return;
  int c   = idx % (3 * Dv);
  int ms  = idx / (3 * Dv);
  int s   = ms & (Sv - 1);
  int bb  = ms >> 12;
  int d   = c & (Dv - 1);
  int grp = c >> 7;
  const float* w = (grp == 0) ? w0 : ((grp == 1) ? w1 : w2);
  int kk  = (grp == 0) ? k0 : ((grp == 1) ? k1 : k2);
  int pad = kk >> 1;
  float acc = 0.f;
  for (int k = 0; k < kk; ++k) {
    int si = s - pad + k;
    if (si >= 0 && si < Sv)
      acc += src[(((size_t)(bb * Sv + si)) << 7) + d] * w[d * kk + k];
  }
  out[(size_t)ms * (3 * Dv) + c] = acc;
}

// ---------------------------------------------------------------------------
// GEMM + batchnorm + silu + optional residual ; 16x64 per wave
// ---------------------------------------------------------------------------
__global__ void k_gemm_bn_silu_res(const float* A, int lda, const float* W, int ldw,
                                   const float* g, const float* bb,
                                   const float* mm, const float* vv,
                                   const float* res, int ldr,
                                   float* out, int ldc, int outOff,
                                   int nG, int totalWaves, int K) {
  int wid = blockIdx.x * (blockDim.x >> 5) + (threadIdx.x >> 5);
  if (wid >= totalWaves) return;
  int m0 = (wid / nG) << 4;
  int n0 = (wid % nG) << 6;
  v8f acc[4];
  wmma_tileN<4>(A, lda, W, ldw, m0, n0, K, acc);
  int lane = threadIdx.x & 31;
  int mb = m0 + ((lane >> 4) << 3);
  #pragma unroll
  for (int nt = 0; nt < 4; ++nt) {
    int nn = n0 + nt * 16 + (lane & 15);
    float sc = rsqrtf(vv[nn] + 1e-5f) * g[nn];
    float sh = bb[nn] - mm[nn] * sc;
    #pragma unroll
    for (int v = 0; v < 8; ++v) {
      int m = mb + v;
      float t = siluf(acc[nt][v] * sc + sh);
      if (res) t += res[(size_t)m * ldr + nn];
      out[(size_t)m * ldc + outOff + nn] = t;
    }
  }
}

// GEMM + bias (proj = x @ W_in^T + b_in)
__global__ void k_gemm_bias(const float* A, int lda, const float* W, int ldw,
                            const float* bias, float* out, int ldc,
                            int nG, int totalWaves, int K) {
  int wid = blockIdx.x * (blockDim.x >> 5) + (threadIdx.x >> 5);
  if (wid >= totalWaves) return;
  int m0 = (wid / nG) << 4;
  int n0 = (wid % nG) << 6;
  v8f acc[4];
  wmma_tileN<4>(A, lda, W, ldw, m0, n0, K, acc);
  int lane = threadIdx.x & 31;
  int mb = m0 + ((lane >> 4) << 3);
  #pragma unroll
  for (int nt = 0; nt < 4; ++nt) {
    int nn = n0 + nt * 16 + (lane & 15);
    float bi = bias[nn];
    #pragma unroll
    for (int v = 0; v < 8; ++v)
      out[(size_t)(mb + v) * ldc + nn] = acc[nt][v] + bi;
  }
}

// fused = p1 @ w_p2^T + 0.5*(slow_ctx + fast_ctx)      (cat holds [slow|fast])
__global__ void k_gemm_p2(const float* A, const float* W, const float* cat,
                          float* out, int nG, int totalWaves) {
  int wid = blockIdx.x * (blockDim.x >> 5) + (threadIdx.x >> 5);
  if (wid >= totalWaves) return;
  int m0 = (wid / nG) << 4;
  int n0 = (wid % nG) << 6;
  v8f acc[4];
  wmma_tileN<4>(A, 2 * Dv, W, 2 * Dv, m0, n0, 2 * Dv, acc);
  int lane = threadIdx.x & 31;
  int mb = m0 + ((lane >> 4) << 3);
  #pragma unroll
  for (int nt = 0; nt < 4; ++nt) {
    int nn = n0 + nt * 16 + (lane & 15);
    #pragma unroll
    for (int v = 0; v < 8; ++v) {
      int m = mb + v;
      out[(size_t)m * Dv + nn] = acc[nt][v] +
          0.5f * (cat[(size_t)m * 2 * Dv + nn] + cat[(size_t)m * 2 * Dv + Dv + nn]);
    }
  }
}

// ---------------------------------------------------------------------------
// TDM-staged modulation GEMMs (K=128, N=256): all 8 waves of a block share a
// 64-out-channel weight panel; one wave DMAs it to LDS via the Tensor Data
// Mover, then WMMA reads W through ds_load.
// ---------------------------------------------------------------------------
// u = proj[:, :256] * (1 + sigmoid(nc @ W_ids^T + b_ids))
__global__ void __launch_bounds__(256)
k_gemm_drive(const float* __restrict__ nc, const float* __restrict__ W,
             const float* __restrict__ bias, const float* __restrict__ proj,
             float* __restrict__ u) {
  __shared__ float wlds[64 * Dv];                 // 32 KB weight panel
  const int nG = INNERv / 64;                     // 4 panels
  int mBlk = blockIdx.x / nG;
  int ng   = blockIdx.x % nG;
  if ((threadIdx.x >> 5) == 0)
    tdm_stage_rows(W + (size_t)ng * 64 * Dv, wlds, Dv, 64);
  __syncthreads();
  int wave = threadIdx.x >> 5;
  int m0 = mBlk * 128 + wave * 16;
  v8f acc[4];
  wmma_tileN<4>(nc, Dv, wlds, Dv, m0, 0, Dv, acc);
  int lane = threadIdx.x & 31;
  int mb = m0 + ((lane >> 4) << 3);
  #pragma unroll
  for (int nt = 0; nt < 4; ++nt) {
    int nn = ng * 64 + nt * 16 + (lane & 15);
    float bi = bias[nn];
    #pragma unroll
    for (int v = 0; v < 8; ++v) {
      int m = mb + v;
      u[(size_t)m * INNERv + nn] =
          proj[(size_t)m * (4 * Dv) + nn] * (1.f + sigmf(acc[nt][v] + bi));
    }
  }
}

// gate = proj[:, 256:] + nc @ W_rgb^T + b_rgb
__global__ void __launch_bounds__(256)
k_gemm_gate(const float* __restrict__ nc, const float* __restrict__ W,
            const float* __restrict__ bias, const float* __restrict__ proj,
            float* __restrict__ gate) {
  __shared__ float wlds[64 * Dv];
  const int nG = INNERv / 64;
  int mBlk = blockIdx.x / nG;
  int ng   = blockIdx.x % nG;
  if ((threadIdx.x >> 5) == 0)
    tdm_stage_rows(W + (size_t)ng * 64 * Dv, wlds, Dv, 64);
  __syncthreads();
  int wave = threadIdx.x >> 5;
  int m0 = mBlk * 128 + wave * 16;
  v8f acc[4];
  wmma_tileN<4>(nc, Dv, wlds, Dv, m0, 0, Dv, acc);
  int lane = threadIdx.x & 31;
  int mb = m0 + ((lane >> 4) << 3);
  #pragma unroll
  for (int nt = 0; nt < 4; ++nt) {
    int nn = ng * 64 + nt * 16 + (lane & 15);
    float bi = bias[nn];
    #pragma unroll
    for (int v = 0; v < 8; ++v) {
      int m = mb + v;
      gate[(size_t)m * INNERv + nn] =
          proj[(size_t)m * (4 * Dv) + 2 * Dv + nn] + acc[nt][v] + bi;
    }
  }
}

// d_out = (y*silu(gate)) @ W_out^T + b_out + tokens
__global__ void k_gemm_out(const float* A, const float* W, const float* bias,
                           const float* tokens, float* out,
                           int nG, int totalWaves) {
  int wid = blockIdx.x * (blockDim.x >> 5) + (threadIdx.x >> 5);
  if (wid >= totalWaves) return;
  int m0 = (wid / nG) << 4;
  int n0 = (wid % nG) << 6;
  v8f acc[4];
  wmma_tileN<4>(A, INNERv, W, INNERv, m0, n0, INNERv, acc);
  int lane = threadIdx.x & 31;
  int mb = m0 + ((lane >> 4) << 3);
  #pragma unroll
  for (int nt = 0; nt < 4; ++nt) {
    int nn = n0 + nt * 16 + (lane & 15);
    float bi = bias[nn];
    #pragma unroll
    for (int v = 0; v < 8; ++v) {
      int m = mb + v;
      out[(size_t)m * Dv + nn] = acc[nt][v] + bi + tokens[(size_t)m * Dv + nn];
    }
  }
}

// causal depthwise conv k=3 (pad 2, truncated) + bias + silu   [B*S, 256]
__global__ void k_causal_silu(const float* __restrict__ u, const float* __restrict__ w,
                              const float* __restrict__ bl, float* __restrict__ out) {
  int idx = blockIdx.x * blockDim.x + threadIdx.x;
  if (idx >= Mrows * INNERv) return;
  int c  = idx & (INNERv - 1);
  int ms = idx >> 8;
  int s  = ms & (Sv - 1);
  int bb = ms >> 12;
  float acc = bl[c];
  #pragma unroll
  for (int k = 0; k < 3; ++k) {
    int si = s - 2 + k;
    if (si >= 0)
      acc += u[(((size_t)(bb * Sv + si)) << 8) + c] * w[c * 3 + k];
  }
  out[idx] = siluf(acc);
}

// sp = u2 @ W_sp^T   (N=40 not a WMMA multiple -> VALU)
__global__ void k_sp(const float* __restrict__ u2, const float* __restrict__ Wsp,
                     float* __restrict__ sp) {
  int idx = blockIdx.x * blockDim.x + threadIdx.x;
  if (idx >= Mrows * 40) return;
  int j = idx % 40;
  int m = idx / 40;
  const float* ur = u2 + (size_t)m * INNERv;
  const float* wr = Wsp + (size_t)j * INNERv;
  float acc = 0.f;
  for (int i = 0; i < INNERv; ++i) acc += ur[i] * wr[i];
  sp[idx] = acc;
}

// delta = softplus(sp[:, :8] @ W_dp^T + b_dp)
__global__ void k_delta(const float* __restrict__ sp, const float* __restrict__ Wdp,
                        const float* __restrict__ bdp, float* __restrict__ delta) {
  int idx = blockIdx.x * blockDim.x + threadIdx.x;
  if (idx >= Mrows * INNERv) return;
  int o = idx & (INNERv - 1);
  int m = idx >> 8;
  float x = bdp[o];
  #pragma unroll
  for (int k = 0; k < DRv; ++k) x += sp[(size_t)m * 40 + k] * Wdp[o * DRv + k];
  delta[idx] = (x > 20.f) ? x : log1pf(expf(x));
}

// ---------------------------------------------------------------------------
// selective scan: one lane per (channel, n) state; 16-lane shfl_xor reduction
// grid = B * INNER/16 = 32 blocks, 256 threads
// ---------------------------------------------------------------------------
__global__ void k_scan(const float* __restrict__ delta, const float* __restrict__ u2,
                       const float* __restrict__ sp, const float* __restrict__ A_log,
                       const float* __restrict__ Dsk, float* __restrict__ y) {
  int bb    = blockIdx.x >> 4;
  int cBase = (blockIdx.x & 15) << 4;
  int c = cBase + (threadIdx.x >> 4);
  int n = threadIdx.x & 15;
  float Ac  = -expf(A_log[c * Nv + n]);
  float dsk = Dsk[c];
  float h = 0.f;
  size_t base = (size_t)bb * Sv;
  for (int t = 0; t < Sv; ++t) {
    size_t rowi = base + t;
    float dlt = delta[rowi * INNERv + c];
    float uu  = u2[rowi * INNERv + c];
    float Bn  = sp[rowi * 40 + DRv + n];
    float Cn  = sp[rowi * 40 + DRv + Nv + n];
    h = expf(dlt * Ac) * h + dlt * Bn * uu;
    float contrib = h * Cn;
    #pragma unroll
    for (int off = 8; off; off >>= 1) contrib += __shfl_xor(contrib, off, 16);
    if (n == 0) y[rowi * INNERv + c] = contrib + uu * dsk;
  }
}

// y *= silu(gate)   (in place)
__global__ void k_ygate(float* __restrict__ y, const float* __restrict__ gate) {
  int idx = blockIdx.x * blockDim.x + threadIdx.x;
  if (idx >= Mrows * INNERv) return;
  y[idx] = y[idx] * siluf(gate[idx]);
}

// ---------------------------------------------------------------------------
extern "C" void kernel_launch(void* const* d_in, const int* in_sizes, int n_in,
                              void* d_out, int out_size, void* d_ws, size_t ws_size,
                              hipStream_t stream) {
  (void)in_sizes; (void)n_in; (void)out_size; (void)ws_size;
  const float* tokens   = (const float*)d_in[0];
  const float* ln_g     = (const float*)d_in[1];
  const float* ln_b     = (const float*)d_in[2];
  const float* w_low    = (const float*)d_in[3];
  const float* w_s5     = (const float*)d_in[4];
  const float* w_s9     = (const float*)d_in[5];
  const float* w_s13    = (const float*)d_in[6];
  const float* w_f3     = (const float*)d_in[7];
  const float* w_f5     = (const float*)d_in[8];
  const float* w_f7     = (const float*)d_in[9];
  const float* w_sf     = (const float*)d_in[10];
  const float* sf_g     = (const float*)d_in[11];
  const float* sf_b     = (const float*)d_in[12];
  const float* sf_m     = (const float*)d_in[13];
  const float* sf_v     = (const float*)d_in[14];
  const float* w_ff     = (const float*)d_in[15];
  const float* ff_g     = (const float*)d_in[16];
  const float* ff_b     = (const float*)d_in[17];
  const float* ff_m     = (const float*)d_in[18];
  const float* ff_v     = (const float*)d_in[19];
  const float* w_p1     = (const float*)d_in[20];
  const float* p_g      = (const float*)d_in[21];
  const float* p_b      = (const float*)d_in[22];
  const float* p_m      = (const float*)d_in[23];
  const float* p_v      = (const float*)d_in[24];
  const float* w_p2     = (const float*)d_in[25];
  const float* ctx_ln_g = (const float*)d_in[26];
  const float* ctx_ln_b = (const float*)d_in[27];
  const float* cm_ln_g  = (const float*)d_in[28];
  const float* cm_ln_b  = (const float*)d_in[29];
  const float* W_ids    = (const float*)d_in[30];
  const float* b_ids    = (const float*)d_in[31];
  const float* W_rgb    = (const float*)d_in[32];
  const float* b_rgb    = (const float*)d_in[33];
  const float* W_in     = (const float*)d_in[34];
  const float* b_in     = (const float*)d_in[35];
  const float* w_lt     = (const float*)d_in[36];
  const float* b_lt     = (const float*)d_in[37];
  const float* W_sp     = (const float*)d_in[38];
  const float* W_dp     = (const float*)d_in[39];
  const float* b_dp     = (const float*)d_in[40];
  const float* A_log    = (const float*)d_in[41];
  const float* D_skip   = (const float*)d_in[42];
  const float* W_out    = (const float*)d_in[43];
  const float* b_out    = (const float*)d_in[44];

  // workspace carve-out (floats); ~88 MB total with reuse
  float* ws    = (float*)d_ws;
  float* x     = ws;                                 // M*128
  float* proj  = x     + (size_t)Mrows * Dv;         // M*512
  float* slow  = proj  + (size_t)Mrows * 4 * Dv;     // M*128
  float* fast  = slow  + (size_t)Mrows * Dv;         // M*128
  float* msb   = fast  + (size_t)Mrows * Dv;         // M*384  (reused as u2)
  float* catb  = msb   + (size_t)Mrows * 3 * Dv;     // M*256  (reused as y)
  float* p1b   = catb  + (size_t)Mrows * 2 * Dv;     // M*256  (reused as delta)
  float* fused = p1b   + (size_t)Mrows * 2 * Dv;     // M*128
  float* ctx   = fused + (size_t)Mrows * Dv;         // M*128
  float* ncb   = ctx   + (size_t)Mrows * Dv;         // M*128
  float* ubuf  = ncb   + (size_t)Mrows * Dv;         // M*256
  float* gate  = ubuf  + (size_t)Mrows * INNERv;     // M*256
  float* u2    = msb;    // free after fast_ctx GEMM
  float* spb   = fast;   // free after fast_ctx GEMM (M*40 <= M*128)
  float* dlb   = p1b;    // free after p2 GEMM
  float* yb    = catb;   // free after p2 GEMM

  const int mT = Mrows / 16;                         // 512 m-tiles
  auto eg = [](long n) { return dim3((unsigned)((n + 255) / 256)); };
  auto gw = [](int waves) { return dim3((unsigned)((waves + 7) / 8)); };
  dim3 blk(256);

  // 1. x = LN(tokens)
  k_ln<<<dim3(Mrows / 8), blk, 0, stream>>>(tokens, ln_g, ln_b, x, Mrows);
  // 2. proj = x @ W_in^T + b_in           (K=128, N=512 -> 8 n-groups)
  k_gemm_bias<<<gw(mT * 8), blk, 0, stream>>>(x, Dv, W_in, Dv, b_in, proj, 4 * Dv,
                                              8, mT * 8, Dv);
  // 3. slow / fast
  k_dwlow<<<eg((long)Mrows * Dv), blk, 0, stream>>>(x, w_low, slow, fast);
  // 4. slow_ms -> GEMM(w_sf)+BN+silu+slow -> cat[:, :128]   (K=384, N=128 -> 2)
  k_msconv<<<eg((long)Mrows * 3 * Dv), blk, 0, stream>>>(slow, w_s5, 5, w_s9, 9,
                                                         w_s13, 13, msb);
  k_gemm_bn_silu_res<<<gw(mT * 2), blk, 0, stream>>>(msb, 3 * Dv, w_sf, 3 * Dv,
      sf_g, sf_b, sf_m, sf_v, slow, Dv, catb, 2 * Dv, 0, 2, mT * 2, 3 * Dv);
  // 5. fast_ms -> GEMM(w_ff)+BN+silu+fast -> cat[:, 128:]
  k_msconv<<<eg((long)Mrows * 3 * Dv), blk, 0, stream>>>(fast, w_f3, 3, w_f5, 5,
                                                         w_f7, 7, msb);
  k_gemm_bn_silu_res<<<gw(mT * 2), blk, 0, stream>>>(msb, 3 * Dv, w_ff, 3 * Dv,
      ff_g, ff_b, ff_m, ff_v, fast, Dv, catb, 2 * Dv, Dv, 2, mT * 2, 3 * Dv);
  // 6. p1 = silu(BN(cat @ w_p1^T))        (K=256, N=256 -> 4)
  k_gemm_bn_silu_res<<<gw(mT * 4), blk, 0, stream>>>(catb, 2 * Dv, w_p1, 2 * Dv,
      p_g, p_b, p_m, p_v, nullptr, 0, p1b, 2 * Dv, 0, 4, mT * 4, 2 * Dv);
  // 7. fused = p1 @ w_p2^T + 0.5*(slow_ctx + fast_ctx)      (N=128 -> 2)
  k_gemm_p2<<<gw(mT * 2), blk, 0, stream>>>(p1b, w_p2, catb, fused, 2, mT * 2);
  // 8. ctx = LN(fused); nc = LN(ctx)
  k_ln<<<dim3(Mrows / 8), blk, 0, stream>>>(fused, ctx_ln_g, ctx_ln_b, ctx, Mrows);
  k_ln<<<dim3(Mrows / 8), blk, 0, stream>>>(ctx, cm_ln_g, cm_ln_b, ncb, Mrows);
  // 9. u / gate modulation GEMMs via TDM-staged LDS weight panels
  //    grid = (M/128 mBlks) * (256/64 panels) = 64*4 = 256 blocks
  k_gemm_drive<<<dim3((Mrows / 128) * (INNERv / 64)), blk, 0, stream>>>(
      ncb, W_ids, b_ids, proj, ubuf);
  k_gemm_gate<<<dim3((Mrows / 128) * (INNERv / 64)), blk, 0, stream>>>(
      ncb, W_rgb, b_rgb, proj, gate);
  // 10. causal conv + silu -> u2
  k_causal_silu<<<eg((long)Mrows * INNERv), blk, 0, stream>>>(ubuf, w_lt, b_lt, u2);
  // 11. sp, delta
  k_sp<<<eg((long)Mrows * 40), blk, 0, stream>>>(u2, W_sp, spb);
  k_delta<<<eg((long)Mrows * INNERv), blk, 0, stream>>>(spb, W_dp, b_dp, dlb);
  // 12. selective scan -> y
  k_scan<<<dim3(32), blk, 0, stream>>>(dlb, u2, spb, A_log, D_skip, yb);
  // 13. y *= silu(gate); out = y @ W_out^T + b_out + tokens  (N=128 -> 2)
  k_ygate<<<eg((long)Mrows * INNERv), blk, 0, stream>>>(yb, gate);
  k_gemm_out<<<gw(mT * 2), blk, 0, stream>>>(yb, W_out, b_out, tokens,
                                             (float*)d_out, 2, mT * 2);
}